// SegmentedPolynomialFromUniform1dJit_46497315947083
// MI455X (gfx1250) — compile-verified
//
#include <hip/hip_runtime.h>

// Problem constants (from reference): E=128, S=64, P=256, SE = S*E = 8192.
#define E_DIM    128
#define S_DIM    64
#define P_DIM    256
#define SE       8192
#define NTHREADS 256
// Padded row stride (floats) for the Z^T tile in LDS: even (keeps 8B alignment
// for k-pair b64 loads) and != 0 mod 64 banks (kills write conflicts).
#define ZSTR     258

#if __has_builtin(__builtin_amdgcn_global_load_async_to_lds_b128)
#define HAVE_ASYNC_LDS 1
#else
#define HAVE_ASYNC_LDS 0
#endif

typedef float v2f __attribute__((ext_vector_type(2)));
typedef float v8f __attribute__((ext_vector_type(8)));
typedef int   v4i_t __attribute__((vector_size(4 * sizeof(int))));

// out[b] (64x128) = M (64x256, M[s,p] = c_p * [s3(p)==s]) x Z_b (256x128,
// Z_b[p,e] = x_table[x_idx[b]][s1(p)*128+e] * y[b][s2(p)*128+e])
__global__ __launch_bounds__(NTHREADS)
void segtp_u1d_wmma_f32_kernel(const float* __restrict__ x_table,
                               const float* __restrict__ y,
                               const float* __restrict__ path_coeff,
                               const int*   __restrict__ x_idx,
                               const int*   __restrict__ path_idx,
                               float* __restrict__ out)
{
  extern __shared__ float smem[];
  float* s_x  = smem;                       // 8192 f32 : gathered x row
  float* s_y  = s_x + SE;                   // 8192 f32 : y row
  int*   s12  = (int*)(s_y + SE);           // 512 i32  : {s1,s2} per path
  int*   s_sc = s12 + 2 * P_DIM;            // 512 i32  : {s3, bits(c)} per path
  float* s_z  = (float*)(s_sc + 2 * P_DIM); // 128 x ZSTR f32 : Z^T[e][p]

  const int tid = threadIdx.x;
  const int b   = blockIdx.x;

  // ---- Stage x (gathered row) and y into LDS ----
  // CDNA5 path: cache->LDS async copies (ASYNCcnt), no VGPR staging; the
  // path-table setup below overlaps with the copies.
  {
    const int row = x_idx[b];
    const float4* xr = (const float4*)(x_table + (size_t)row * SE);
    const float4* yr = (const float4*)(y + (size_t)b * SE);
    float4* sx4 = (float4*)s_x;
    float4* sy4 = (float4*)s_y;
#if HAVE_ASYNC_LDS
#pragma unroll
    for (int i = 0; i < (SE / 4) / NTHREADS; ++i) {
      const int j = tid + i * NTHREADS;
      __builtin_amdgcn_global_load_async_to_lds_b128(
          (v4i_t*)(xr + j), (v4i_t*)(sx4 + j), 0, 0);
      __builtin_amdgcn_global_load_async_to_lds_b128(
          (v4i_t*)(yr + j), (v4i_t*)(sy4 + j), 0, 0);
    }
#else
#pragma unroll
    for (int i = 0; i < (SE / 4) / NTHREADS; ++i) {
      const int j = tid + i * NTHREADS;
      sx4[j] = xr[j];
      sy4[j] = yr[j];
    }
#endif
  }

  // ---- Stage path tables (256 threads == P paths, one each) ----
  {
    const int p  = tid;
    const int i1 = path_idx[3 * p + 0];
    const int i2 = path_idx[3 * p + 1];
    const int i3 = path_idx[3 * p + 2];
    const float c = path_coeff[p];
    s12[2 * p + 0]  = i1;
    s12[2 * p + 1]  = i2;
    s_sc[2 * p + 0] = i3;
    s_sc[2 * p + 1] = __float_as_int(c);
  }

#if HAVE_ASYNC_LDS
  // Wait for this wave's async cache->LDS copies to land before the barrier.
#if __has_builtin(__builtin_amdgcn_s_wait_asynccnt)
  __builtin_amdgcn_s_wait_asynccnt(0);
#else
  asm volatile("s_wait_asynccnt 0" ::: "memory");
#endif
#endif
  __syncthreads();

  // ---- Build Z^T[e][p] = x[s1(p)][e] * y[s2(p)][e] ----
  // Thread owns one e (lanes -> consecutive e, so x/y reads are conflict-free
  // despite the 128-float segment stride) and half of the p range.
  {
    const int e     = tid & (E_DIM - 1);
    const int pbase = (tid >> 7) * (P_DIM / 2);
#pragma unroll 4
    for (int pp = 0; pp < P_DIM / 2; ++pp) {
      const int p  = pbase + pp;
      const int i1 = s12[2 * p + 0];   // broadcast read (same p across lanes)
      const int i2 = s12[2 * p + 1];
      s_z[e * ZSTR + p] = s_x[i1 * E_DIM + e] * s_y[i2 * E_DIM + e];
    }
  }
  __syncthreads();

  // ---- WMMA stage: wave w owns e-columns [16w, 16w+16) for all 4 s-tiles ----
  const int wave = tid >> 5;
  const int lane = tid & 31;
  const int row  = lane & 15;   // A-matrix row (M) == B/D column (N)
  const int half = lane >> 4;   // K-half selector per ISA 16x4 A layout

  v8f acc[4] = {};              // four 16x16 f32 accumulators (s-tiles 0..3)
  const float* zcol = s_z + (wave * 16 + row) * ZSTR;

  for (int k = 0; k < P_DIM; k += 4) {
    const int k0 = k + half * 2;                 // this lane's K pair start
    // {s3,c} for k0 and k0+1: one b128 broadcast load (same addr per half)
    const int4 sc = *(const int4*)&s_sc[2 * k0];
    const int   s3a = sc.x;  const float ca = __int_as_float(sc.y);
    const int   s3b = sc.z;  const float cb = __int_as_float(sc.w);
    // B fragment: Z[k0][n], Z[k0+1][n] -> contiguous pair in Z^T row (b64)
    const v2f bfrag = *(const v2f*)&zcol[k0];
#pragma unroll
    for (int st = 0; st < 4; ++st) {
      const int tgt = st * 16 + row;
      v2f afrag;
      afrag.x = (s3a == tgt) ? ca : 0.0f;        // A[row][k0]
      afrag.y = (s3b == tgt) ? cb : 0.0f;        // A[row][k0+1]
      acc[st] = __builtin_amdgcn_wmma_f32_16x16x4_f32(
          /*neg_a=*/false, afrag, /*neg_b=*/false, bfrag,
          /*c_mod=*/(short)0, acc[st], /*reuse_a=*/false, /*reuse_b=*/false);
    }
  }

  // ---- Store D tiles: VGPR r holds rows r (lanes 0-15) / r+8 (lanes 16-31).
  // Output is write-once streaming: non-temporal so it doesn't evict
  // gather-reusable x_table lines from the 192MB L2.
  float* ob = out + (size_t)b * SE + wave * 16 + row;
#pragma unroll
  for (int st = 0; st < 4; ++st) {
#pragma unroll
    for (int r = 0; r < 8; ++r) {
      const int m = st * 16 + half * 8 + r;
      __builtin_nontemporal_store(acc[st][r], &ob[m * E_DIM]);
    }
  }
}

extern "C" void kernel_launch(void* const* d_in, const int* in_sizes, int n_in,
                              void* d_out, int out_size, void* d_ws, size_t ws_size,
                              hipStream_t stream) {
  (void)in_sizes; (void)n_in; (void)d_ws; (void)ws_size;
  const float* x_table    = (const float*)d_in[0];
  const float* yv         = (const float*)d_in[1];
  const float* path_coeff = (const float*)d_in[2];
  const int*   x_idx      = (const int*)d_in[3];
  const int*   path_idx   = (const int*)d_in[4];
  float* out = (float*)d_out;

  const int batches = out_size / SE;  // 4096
  const size_t smem_bytes =
      (size_t)(SE + SE + 2 * P_DIM + 2 * P_DIM + E_DIM * ZSTR) * sizeof(float);

  segtp_u1d_wmma_f32_kernel<<<batches, NTHREADS, smem_bytes, stream>>>(
      x_table, yv, path_coeff, x_idx, path_idx, out);
}